// GuidedFilterHR_34196529610984
// MI455X (gfx1250) — compile-verified
//
#include <hip/hip_runtime.h>
#include <hip/hip_bf16.h>
#include <math.h>

typedef __attribute__((ext_vector_type(2))) float v2f;
typedef __attribute__((ext_vector_type(8))) float v8f;
typedef __attribute__((ext_vector_type(4))) unsigned int v4u;
typedef __attribute__((ext_vector_type(8))) int v8i;
typedef __attribute__((ext_vector_type(4))) int v4i;

#define IMG 512
#define NPIX (IMG * IMG)

#if defined(__gfx1250__) && __has_builtin(__builtin_amdgcn_tensor_load_to_lds) && \
    __has_builtin(__builtin_amdgcn_s_wait_tensorcnt)
#define USE_TDM 1
#else
#define USE_TDM 0
#endif

// ---------------------------------------------------------------------------
// Kernel 1: block-level min/max partial reduction of y
// ---------------------------------------------------------------------------
__global__ void gf_reduce_minmax(const float* __restrict__ y,
                                 float* __restrict__ pmin,
                                 float* __restrict__ pmax) {
    __shared__ float smin[256];
    __shared__ float smax[256];
    int tid = threadIdx.x;
    int gid = blockIdx.x * 256 + tid;
    float mn = 1e30f, mx = -1e30f;
    for (int idx = gid; idx < NPIX; idx += 256 * 256) {
        float v = y[idx];
        mn = fminf(mn, v);
        mx = fmaxf(mx, v);
    }
    smin[tid] = mn; smax[tid] = mx;
    __syncthreads();
    for (int off = 128; off > 0; off >>= 1) {
        if (tid < off) {
            smin[tid] = fminf(smin[tid], smin[tid + off]);
            smax[tid] = fmaxf(smax[tid], smax[tid + off]);
        }
        __syncthreads();
    }
    if (tid == 0) { pmin[blockIdx.x] = smin[0]; pmax[blockIdx.x] = smax[0]; }
}

// ---------------------------------------------------------------------------
// Kernel 2: final reduction + sigma = r * (max - min)
// ---------------------------------------------------------------------------
__global__ void gf_finalize_sigma(const float* __restrict__ pmin,
                                  const float* __restrict__ pmax,
                                  const float* __restrict__ r,
                                  float* __restrict__ sig) {
    __shared__ float smin[256];
    __shared__ float smax[256];
    int tid = threadIdx.x;
    smin[tid] = pmin[tid]; smax[tid] = pmax[tid];
    __syncthreads();
    for (int off = 128; off > 0; off >>= 1) {
        if (tid < off) {
            smin[tid] = fminf(smin[tid], smin[tid + off]);
            smax[tid] = fmaxf(smax[tid], smax[tid + off]);
        }
        __syncthreads();
    }
    if (tid == 0) sig[0] = r[0] * (smax[0] - smin[0]);
}

// ---------------------------------------------------------------------------
// Kernel 3: horizontal 9-tap mean blur as banded GEMM:  H = Img * S
// S[k,n] = 1/9 when |k-n| <= 4 (zero padding at borders for free).
// One wave = one 16x16 output tile; 6x V_WMMA_F32_16X16X4_F32 over the band.
// ---------------------------------------------------------------------------
__global__ void gf_blur_h_wmma(const float* __restrict__ X,
                               const float* __restrict__ Y,
                               float* __restrict__ HX,
                               float* __restrict__ HY) {
    int wave = blockIdx.x * 8 + (threadIdx.x >> 5);
    int lane = threadIdx.x & 31;
    int img  = wave >> 10;              // 1024 tiles per image
    int t    = wave & 1023;
    int m0   = (t >> 5) << 4;
    int n0   = (t & 31) << 4;
    const float* src = img ? Y : X;
    float* dst       = img ? HY : HX;

    int half = lane >> 4;               // lane half selects K pair
    int l15  = lane & 15;
    int mA   = m0 + l15;                // A-matrix row owned by this lane

    v8f acc = {0.f, 0.f, 0.f, 0.f, 0.f, 0.f, 0.f, 0.f};
    #pragma unroll
    for (int seg = 0; seg < 6; ++seg) {
        int k0 = n0 - 4 + 4 * seg;      // band covers columns n0-4 .. n0+19
        int ka = k0 + (half << 1);
        v2f a;
        a.x = (ka     >= 0 && ka     < IMG) ? src[mA * IMG + ka]     : 0.f;
        a.y = (ka + 1 >= 0 && ka + 1 < IMG) ? src[mA * IMG + ka + 1] : 0.f;
        // banded constant B: B[kk][nn] = (|k - n| <= 4) * 1/9
        int d0 = (k0 + (half << 1)) - (n0 + l15);
        v2f b;
        b.x = (d0     >= -4 && d0     <= 4) ? (1.0f / 9.0f) : 0.f;
        b.y = (d0 + 1 >= -4 && d0 + 1 <= 4) ? (1.0f / 9.0f) : 0.f;
        acc = __builtin_amdgcn_wmma_f32_16x16x4_f32(
            false, a, false, b, (short)0, acc, false, false);
    }
    // D layout: VGPR r -> (M = r + 8*half, N = l15)
    int col = n0 + l15;
    int rowBase = m0 + half * 8;
    #pragma unroll
    for (int rr = 0; rr < 8; ++rr)
        dst[(rowBase + rr) * IMG + col] = acc[rr];
}

// ---------------------------------------------------------------------------
// Kernel 4: vertical 9-tap mean blur as banded GEMM:  Base = S * H
// ---------------------------------------------------------------------------
__global__ void gf_blur_v_wmma(const float* __restrict__ HX,
                               const float* __restrict__ HY,
                               float* __restrict__ BX,
                               float* __restrict__ BY) {
    int wave = blockIdx.x * 8 + (threadIdx.x >> 5);
    int lane = threadIdx.x & 31;
    int img  = wave >> 10;
    int t    = wave & 1023;
    int m0   = (t >> 5) << 4;
    int n0   = (t & 31) << 4;
    const float* src = img ? HY : HX;
    float* dst       = img ? BY : BX;

    int half = lane >> 4;
    int l15  = lane & 15;
    int mA   = m0 + l15;

    v8f acc = {0.f, 0.f, 0.f, 0.f, 0.f, 0.f, 0.f, 0.f};
    #pragma unroll
    for (int seg = 0; seg < 6; ++seg) {
        int k0 = m0 - 4 + 4 * seg;      // band covers rows m0-4 .. m0+19
        int ka = k0 + (half << 1);
        int d0 = mA - ka;               // A[mm][kk] banded constant
        v2f a;
        a.x = (d0     >= -4 && d0     <= 4) ? (1.0f / 9.0f) : 0.f;
        a.y = (d0 - 1 >= -4 && d0 - 1 <= 4) ? (1.0f / 9.0f) : 0.f;
        int nB = n0 + l15;
        v2f b;
        b.x = (ka     >= 0 && ka     < IMG) ? src[(ka)     * IMG + nB] : 0.f;
        b.y = (ka + 1 >= 0 && ka + 1 < IMG) ? src[(ka + 1) * IMG + nB] : 0.f;
        acc = __builtin_amdgcn_wmma_f32_16x16x4_f32(
            false, a, false, b, (short)0, acc, false, false);
    }
    int col = n0 + l15;
    int rowBase = m0 + half * 8;
    #pragma unroll
    for (int rr = 0; rr < 8; ++rr)
        dst[(rowBase + rr) * IMG + col] = acc[rr];
}

#if USE_TDM
// Build TDM descriptor group0: count=1, lds_addr, 57-bit global addr, type=2
__device__ __forceinline__ v4u tdm_group0(unsigned lds_addr, unsigned long long gaddr) {
    v4u g;
    g.x = 1u;                                    // count=1, is_restore=0
    g.y = lds_addr;                              // lds_addr [63:32]
    g.z = (unsigned)(gaddr & 0xFFFFFFFFull);     // global_addr [95:64]
    g.w = (unsigned)((gaddr >> 32) & 0x1FFFFFFull) | (2u << 30); // addr hi + type=2
    return g;
}

// group1: data_size=4B, tensor_dim0=512, tensor_dim1=nrows, tile {W, nrows},
// tensor_dim0_stride = 512 elements. 2-D tile -> higher dims zero.
__device__ __forceinline__ v8i tdm_group1(int W, int nrows) {
    v8i g;
    g[0] = (int)(2u << 16);                      // data_size code 2 = 4 bytes
    g[1] = (int)((IMG & 0xFFFF) << 16);          // tensor_dim0 lo16 in [31:16]
    g[2] = (int)((nrows & 0xFFFF) << 16);        // dim0 hi16=0 | tensor_dim1 lo16
    g[3] = (int)((W & 0xFFFF) << 16);            // dim1 hi16=0 | tile_dim0
    g[4] = (int)(nrows & 0xFFFF);                // tile_dim1 | tile_dim2=0
    g[5] = IMG;                                  // tensor_dim0_stride lo32
    g[6] = 0;                                    // stride0 hi | stride1 lo
    g[7] = 0;                                    // stride1 hi
    return g;
}

// One 2-D strided slab DMA: (nrows x W) from gaddr (row stride 512 floats)
// into LDS at lds_addr, rows packed with stride W.
__device__ __forceinline__ void tdm_load_slab(unsigned lds_addr,
                                              unsigned long long gaddr,
                                              const v8i& g1) {
    v4i gz4 = {0, 0, 0, 0};
    v8i gz8 = {0, 0, 0, 0, 0, 0, 0, 0};
    __builtin_amdgcn_tensor_load_to_lds(tdm_group0(lds_addr, gaddr),
                                        g1, gz4, gz4, gz8, 0);
}
#endif

// ---------------------------------------------------------------------------
// Kernel 5: per-chunk bilateral detail filter (5x9) + bilateral-weighted
// guided filter (19x19) + final composition. Slabs staged to LDS via the
// Tensor Data Mover (one 2-D strided DMA per array), TENSORcnt-tracked.
// Chunk geometry (G=64, O=2, N=512 -> 9 chunks, step 62):
//   s = 62*c, W = min(64, 512-s), lo = (c==0)?0:1, trimHi = (c==8)?0:1
// grid = (32 row-tiles, 9 chunks), block = 256.
// ---------------------------------------------------------------------------
__global__ void gf_guided_chunks(const float* __restrict__ X,
                                 const float* __restrict__ Y,
                                 const float* __restrict__ BX,
                                 const float* __restrict__ BY,
                                 const float* __restrict__ sig,
                                 float* __restrict__ out) {
    // flat slabs, runtime stride W (34 rows x up to 64 cols)
    __shared__ float sXd[34 * 64];   // staged as raw X, converted to detail
    __shared__ float sYd[34 * 64];   // staged as raw y, converted to detail
    __shared__ float sXb[34 * 64];
    __shared__ float sYb[34 * 64];

    const int tid   = threadIdx.x;
    const int chunk = blockIdx.y;
    const int i0    = blockIdx.x * 16;          // first output row of tile
    const int s     = 62 * chunk;
    const int W     = min(64, IMG - s);
    const int lo    = (chunk == 0) ? 0 : 1;
    const int trimH = (chunk == 8) ? 0 : 1;
    const int r0    = i0 - 9;                   // first slab row (may be < 0)
    const int rlo   = max(r0, 0);               // clamped global row range
    const int rhi   = min(r0 + 34, IMG);
    const int nrows = rhi - rlo;
    const int zlo   = rlo - r0;                 // slab rows [0,zlo) are OOB
    const int zhi   = rhi - r0;                 // slab rows [zhi,34) are OOB

#if USE_TDM
    // ---- zero-fill out-of-image slab rows (disjoint from TDM target) ----
    for (int p = tid; p < 34 * W; p += 256) {
        int rr = p / W;
        if (rr < zlo || rr >= zhi) {
            sXd[p] = 0.f; sYd[p] = 0.f; sXb[p] = 0.f; sYb[p] = 0.f;
        }
    }
    // ---- wave 0 issues 4 tensor DMAs: (nrows x W) strided slabs -> LDS ----
    if ((threadIdx.x >> 5) == 0) {
        const unsigned long long off =
            (unsigned long long)(rlo * IMG + s) * sizeof(float);
        const unsigned ldsOff = (unsigned)(zlo * W) * (unsigned)sizeof(float);
        v8i g1 = tdm_group1(W, nrows);
        tdm_load_slab((unsigned)(uintptr_t)sXd + ldsOff,
                      (unsigned long long)(uintptr_t)X + off, g1);
        tdm_load_slab((unsigned)(uintptr_t)sYd + ldsOff,
                      (unsigned long long)(uintptr_t)Y + off, g1);
        tdm_load_slab((unsigned)(uintptr_t)sXb + ldsOff,
                      (unsigned long long)(uintptr_t)BX + off, g1);
        tdm_load_slab((unsigned)(uintptr_t)sYb + ldsOff,
                      (unsigned long long)(uintptr_t)BY + off, g1);
        __builtin_amdgcn_s_wait_tensorcnt(0);
    }
    __syncthreads();
    // ---- convert raw X/y slabs to detail bands in place ----
    for (int p = tid; p < 34 * W; p += 256) {
        int rr = p / W;
        if (rr >= zlo && rr < zhi) {
            sXd[p] = sXd[p] - sXb[p];
            sYd[p] = sYd[p] - sYb[p];
        }
    }
    __syncthreads();
#else
    // ---- fallback: manual staged loads ----
    for (int p = tid; p < 34 * W; p += 256) {
        int rr = p / W, cc = p % W;
        int gr = r0 + rr, gc = s + cc;
        float xb = 0.f, yb = 0.f, xd = 0.f, yd = 0.f;
        if (gr >= 0 && gr < IMG) {
            float xv = X[gr * IMG + gc];
            float yv = Y[gr * IMG + gc];
            xb = BX[gr * IMG + gc];
            yb = BY[gr * IMG + gc];
            xd = xv - xb;
            yd = yv - yb;
        }
        sXb[p] = xb; sYb[p] = yb; sXd[p] = xd; sYd[p] = yd;
    }
    __syncthreads();
#endif

    const float sigma  = sig[0];
    const float inv_s2 = 1.0f / (0.25f * sigma * sigma);        // /(sigma/2)^2
    const float invD2d = 1.0f / (11.25f * 11.25f);              // (5*9/4)^2
    const float invD2b = 1.0f / (90.25f * 90.25f);              // (19*19/4)^2

    const int ownedW = W - lo - trimH;
    for (int p = tid; p < 16 * ownedW; p += 256) {
        const int il = p / ownedW;
        const int j  = lo + p % ownedW;          // chunk-local column
        const int i  = i0 + il;                  // global row
        const int sr = il + 9;                   // slab row of center

        // ---- 5x9 bilateral box filter on (yd - Xd), guided by Xd ----
        const float cXd = sXd[sr * W + j];
        float wsum = 0.f, accd = 0.f;
        for (int dr = -2; dr <= 2; ++dr) {
            int gr = i + dr;
            if (gr < 0 || gr >= IMG) continue;
            int rbase = (sr + dr) * W;
            #pragma unroll
            for (int dc = -4; dc <= 4; ++dc) {
                int c = j + dc;
                if (c < 0 || c >= W) continue;
                float xv = sXd[rbase + c];
                float df = xv - cXd;
                float w = __expf(-df * df * inv_s2
                                 - (float)(dr * dr + dc * dc) * invD2d);
                wsum += w;
                accd += w * (sYd[rbase + c] - xv);
            }
        }
        const float bdet = accd / wsum;

        // ---- 19x19 bilateral-weighted guided filter on base band ----
        const float cXb = sXb[sr * W + j];
        float w2 = 0.f, Sy = 0.f, Sx = 0.f, Sxx = 0.f, Sxy = 0.f;
        for (int dr = -9; dr <= 9; ++dr) {
            int gr = i + dr;
            if (gr < 0 || gr >= IMG) continue;
            int rbase = (sr + dr) * W;
            for (int dc = -9; dc <= 9; ++dc) {
                int c = j + dc;
                if (c < 0 || c >= W) continue;
                float xb = sXb[rbase + c];
                float yb = sYb[rbase + c];
                float df = xb - cXb;
                float w = __expf(-df * df * inv_s2
                                 - (float)(dr * dr + dc * dc) * invD2b);
                w2  += w;
                Sy  += w * yb;
                Sx  += w * xb;
                Sxx += w * xb * xb;
                Sxy += w * xb * yb;
            }
        }
        const float inv = 1.0f / w2;
        const float my  = Sy * inv;
        const float mx  = Sx * inv;
        const float var = Sxx * inv - mx * mx;
        const float cov = Sxy * inv - mx * my;
        const float A   = cov / (var + 1e-6f);
        const float b   = my - A * mx;

        out[i * IMG + (s + j)] = A * sXb[sr * W + j] + b + sXd[sr * W + j] + bdet;
    }
}

// ---------------------------------------------------------------------------
// Host launcher
// ---------------------------------------------------------------------------
extern "C" void kernel_launch(void* const* d_in, const int* in_sizes, int n_in,
                              void* d_out, int out_size, void* d_ws, size_t ws_size,
                              hipStream_t stream) {
    const float* X = (const float*)d_in[0];
    const float* Y = (const float*)d_in[1];
    const float* R = (const float*)d_in[2];
    float* out = (float*)d_out;
    float* ws  = (float*)d_ws;

    float* pmin = ws;                 // 256
    float* pmax = ws + 256;           // 256
    float* sig  = ws + 512;           // 1 (slot 512..1023 reserved)
    float* Hx   = ws + 1024;          // 512*512 horizontal blur of X
    float* Hy   = Hx + NPIX;          // 512*512 horizontal blur of y
    float* Bx   = Hy + NPIX;          // 512*512 X base
    float* By   = Bx + NPIX;          // 512*512 y base

    gf_reduce_minmax<<<256, 256, 0, stream>>>(Y, pmin, pmax);
    gf_finalize_sigma<<<1, 256, 0, stream>>>(pmin, pmax, R, sig);
    gf_blur_h_wmma<<<256, 256, 0, stream>>>(X, Y, Hx, Hy);
    gf_blur_v_wmma<<<256, 256, 0, stream>>>(Hx, Hy, Bx, By);
    gf_guided_chunks<<<dim3(32, 9), 256, 0, stream>>>(X, Y, Bx, By, sig, out);
}